// RBAM_Attention_71923522339160
// MI455X (gfx1250) — compile-verified
//
#include <hip/hip_runtime.h>

typedef __attribute__((ext_vector_type(2))) float v2f;
typedef __attribute__((ext_vector_type(4))) float v4f;
typedef __attribute__((ext_vector_type(8))) float v8f;

#define B_  16
#define C_  64
#define D_  16
#define HW_ 2304          // 48*48
#define HID 16
#define NEG_SLOPE 0.01f

// ---------------------------------------------------------------------------
// Kernel 1: global average pool over H,W.  One wave32 per (b,c,d) row.
// 2304 floats/row = 576 float4 = 18 float4 per lane.  Wave32 butterfly reduce.
// Default (RT) loads on purpose: this pass warms the 192MB L2 with x (151MB)
// so the scale kernel's second read of x mostly hits L2.
// ---------------------------------------------------------------------------
__global__ void se_pool_kernel(const float* __restrict__ x,
                               float* __restrict__ pooled) {
    const int wave = threadIdx.x >> 5;                 // 8 waves / block
    const int lane = threadIdx.x & 31;
    const int row  = blockIdx.x * 8 + wave;            // 0 .. 16383
    const v4f* xr = (const v4f*)(x + (size_t)row * HW_);
    float s = 0.0f;
#pragma unroll
    for (int i = 0; i < 18; ++i) {
        v4f v = xr[i * 32 + lane];
        s += v.x + v.y + v.z + v.w;
    }
#pragma unroll
    for (int off = 16; off > 0; off >>= 1)
        s += __shfl_xor(s, off, 32);
    if (lane == 0) pooled[row] = s * (1.0f / (float)HW_);
}

// ---------------------------------------------------------------------------
// Kernel 2: SE gate, single workgroup (16 waves = 512 threads).
// View pooled[B,C,D] as X[(b,d), c] : 256 x 64.  Each wave owns 16 rows.
//   h    = leaky_relu(X @ w1^T + b1)      : 256 x 16   (WMMA f32 16x16x4, K=64)
//   gate = sigmoid(h @ w2^T + b2)         : 256 x 64   (WMMA f32 16x16x4, K=16)
// All fragments are preloaded into register arrays so the 48 loads clause and
// the 16 chained WMMAs issue back-to-back instead of load->wait->wmma x16.
// h is restaged through padded LDS to convert C-layout -> A-layout fragments.
// ---------------------------------------------------------------------------
__global__ void se_gate_kernel(const float* __restrict__ pooled,
                               const float* __restrict__ w1,
                               const float* __restrict__ b1,
                               const float* __restrict__ w2,
                               const float* __restrict__ b2,
                               float* __restrict__ gate) {
    __shared__ float hbuf[16][16][17];                 // [wave][m][k], pad=17

    const int wave = threadIdx.x >> 5;                 // 0..15
    const int lane = threadIdx.x & 31;
    const int n    = lane & 15;                        // N col / A row (M)
    const int hi   = lane >> 4;                        // half-wave select
    const int koff = hi << 1;                          // K offset {0,2}

    // ---- GEMM1: h_tile[16,16] = X_tile[16,64] @ w1^T[64,16] ----
    const int r = (wave << 4) + n;                     // global row (b*16+d)
    const float* xrow = pooled + (r >> 4) * (C_ * D_) + (r & 15); // stride D_ per c

    v2f afrag[16], bfrag[16];
#pragma unroll
    for (int ks = 0; ks < 16; ++ks) {
        const int k = (ks << 2) + koff;
        afrag[ks].x = xrow[k * D_];        afrag[ks].y = xrow[(k + 1) * D_];
        bfrag[ks].x = w1[n * C_ + k];      bfrag[ks].y = w1[n * C_ + k + 1];
    }
    v8f acc = {};
#pragma unroll
    for (int ks = 0; ks < 16; ++ks)
        acc = __builtin_amdgcn_wmma_f32_16x16x4_f32(
                  false, afrag[ks], false, bfrag[ks], (short)0, acc, false, false);

    // bias + leaky relu, C-layout -> LDS [m][k]
    const float bias1 = b1[n];
#pragma unroll
    for (int v = 0; v < 8; ++v) {
        float val = acc[v] + bias1;
        val = (val >= 0.0f) ? val : NEG_SLOPE * val;
        hbuf[wave][v + (hi << 3)][n] = val;
    }
    __syncthreads();

    // ---- GEMM2: gate_tile[16,64] = h_tile[16,16] @ w2^T[16,64] ----
    v2f hfrag[4];
#pragma unroll
    for (int ks = 0; ks < 4; ++ks) {
        const int k = (ks << 2) + koff;
        hfrag[ks].x = hbuf[wave][n][k];
        hfrag[ks].y = hbuf[wave][n][k + 1];
    }
#pragma unroll
    for (int t = 0; t < 4; ++t) {                      // 4 N-tiles of 16
        const int nc = (t << 4) + n;                   // global out channel
        v2f wfrag[4];
#pragma unroll
        for (int ks = 0; ks < 4; ++ks) {
            const int k = (ks << 2) + koff;
            wfrag[ks].x = w2[nc * HID + k];
            wfrag[ks].y = w2[nc * HID + k + 1];
        }
        v8f g = {};
#pragma unroll
        for (int ks = 0; ks < 4; ++ks)
            g = __builtin_amdgcn_wmma_f32_16x16x4_f32(
                    false, hfrag[ks], false, wfrag[ks], (short)0, g, false, false);

        const float bias2 = b2[nc];
#pragma unroll
        for (int v = 0; v < 8; ++v) {
            const int rr = (wave << 4) + v + (hi << 3);    // global row
            float val = g[v] + bias2;
            val = 1.0f / (1.0f + __expf(-val));
            gate[(rr >> 4) * (C_ * D_) + nc * D_ + (rr & 15)] = val;
        }
    }
}

// ---------------------------------------------------------------------------
// Kernel 3: out = x * gate[b,c,d], pure 128-bit stream (1 load + 1 store).
// x read is its last use -> non-temporal load (hits the L2 lines deposited by
// the pool kernel, then releases them).  out store is non-temporal so the
// 151MB output stream does not evict x from the 192MB L2 while we stream.
// Row length 2304 floats = 576 float4.
// ---------------------------------------------------------------------------
__global__ void se_scale_kernel(const float* __restrict__ x,
                                const float* __restrict__ gate,
                                float* __restrict__ out) {
    const size_t idx4 = (size_t)blockIdx.x * blockDim.x + threadIdx.x;
    const v4f* x4 = (const v4f*)x;
    v4f*       o4 = (v4f*)out;
    v4f v = __builtin_nontemporal_load(x4 + idx4);
    const float g = gate[(int)(idx4 / 576)];
    v.x *= g; v.y *= g; v.z *= g; v.w *= g;
    __builtin_nontemporal_store(v, o4 + idx4);
}

// ---------------------------------------------------------------------------
extern "C" void kernel_launch(void* const* d_in, const int* in_sizes, int n_in,
                              void* d_out, int out_size, void* d_ws, size_t ws_size,
                              hipStream_t stream) {
    const float* x  = (const float*)d_in[0];
    const float* w1 = (const float*)d_in[1];
    const float* b1 = (const float*)d_in[2];
    const float* w2 = (const float*)d_in[3];
    const float* b2 = (const float*)d_in[4];
    float* out = (float*)d_out;

    float* pooled = (float*)d_ws;                    // 16384 f32
    float* gate   = pooled + B_ * C_ * D_;           // 16384 f32

    // 16384 rows, 8 waves (rows) per 256-thread block
    se_pool_kernel<<<(B_ * C_ * D_) / 8, 256, 0, stream>>>(x, pooled);

    // single workgroup, 16 waves
    se_gate_kernel<<<1, 512, 0, stream>>>(pooled, w1, b1, w2, b2, gate);

    // 37,748,736 floats = 9,437,184 float4; 256 threads/block -> 36864 blocks
    const size_t total4 = (size_t)B_ * C_ * D_ * HW_ / 4;
    se_scale_kernel<<<(unsigned)(total4 / 256), 256, 0, stream>>>(x, gate, out);
}